// PostProcessor_64098091925954
// MI455X (gfx1250) — compile-verified
//
#include <hip/hip_runtime.h>
#include <hip/hip_bf16.h>
#include <stdint.h>

// Problem shape (compile-time constants from the reference)
#define BB 16
#define NN 512
#define CC 151
#define RR 65536
#define PP 51

#define TILE 2048            // bitonic LDS tile (u64 elements) = 16KB LDS
#define ROWS_PER_BLK 64      // rel rows staged per block
#define TILE_BYTES (ROWS_PER_BLK * PP * 4)   // 13056, multiple of 16
#define CHUNKS (TILE_BYTES / 16)             // 816 x b128 transfers

typedef unsigned int u32;
typedef unsigned long long u64;
typedef int v4i __attribute__((ext_vector_type(4)));

#define GAS __attribute__((address_space(1)))
#define LAS __attribute__((address_space(3)))

#if __has_builtin(__builtin_amdgcn_global_load_async_to_lds_b128)
#define HAVE_ASYNC 1
__device__ __forceinline__ void async_cp16(const void* g, void* l) {
  // signature (from clang diagnostic): param0 = v4i in AS1 (global), then LDS, imm offset, imm cpol
  __builtin_amdgcn_global_load_async_to_lds_b128((GAS v4i*)g, (LAS v4i*)l, 0, 0);
}
#else
#define HAVE_ASYNC 0
#endif

__device__ __forceinline__ void wait_async0() {
#if __has_builtin(__builtin_amdgcn_s_wait_asynccnt)
  __builtin_amdgcn_s_wait_asynccnt(0);
#else
  asm volatile("s_wait_asynccnt 0x0" ::: "memory");
#endif
}

__device__ __forceinline__ float waveMax(float v) {
  #pragma unroll
  for (int off = 16; off > 0; off >>= 1) v = fmaxf(v, __shfl_xor(v, off, 32));
  return v;
}
__device__ __forceinline__ float waveSum(float v) {
  #pragma unroll
  for (int off = 16; off > 0; off >>= 1) v += __shfl_xor(v, off, 32);
  return v;
}
__device__ __forceinline__ void waveArgMax(float& v, int& i) {
  #pragma unroll
  for (int off = 16; off > 0; off >>= 1) {
    float ov = __shfl_xor(v, off, 32);
    int   oi = __shfl_xor(i, off, 32);
    if (ov > v || (ov == v && oi < i)) { v = ov; i = oi; }
  }
}

// float -> u32 key such that ascending u32 order == DESCENDING float order
__device__ __forceinline__ u32 desc_key(float f) {
  u32 b = __float_as_uint(f);
  u32 asc = b ^ ((b & 0x80000000u) ? 0xFFFFFFFFu : 0x80000000u);
  return ~asc;
}

// ---------------------------------------------------------------------------
// Kernel A: obj softmax -> obj_pred (argmax(fg)+1) and obj_scores (max fg prob)
// One wave (32 lanes) per row of 151 logits; 8 waves / block.
// ---------------------------------------------------------------------------
__global__ void obj_softmax_kernel(const float* __restrict__ obj,
                                   float* __restrict__ outPred,
                                   float* __restrict__ outScore) {
  const int wave = threadIdx.x >> 5, lane = threadIdx.x & 31;
  const int row  = blockIdx.x * 8 + wave;            // < B*N
  const float* rp = obj + (size_t)row * CC;
  float v[5]; float m = -INFINITY;
  #pragma unroll
  for (int t = 0; t < 5; ++t) {
    int idx = lane + 32 * t;
    v[t] = (idx < CC) ? rp[idx] : -INFINITY;
    m = fmaxf(m, v[t]);
  }
  m = waveMax(m);
  float s = 0.f;
  #pragma unroll
  for (int t = 0; t < 5; ++t) {
    int idx = lane + 32 * t;
    if (idx < CC) s += __expf(v[t] - m);
  }
  s = waveSum(s);
  // foreground max/argmax over classes 1..150 (softmax is monotone in logits)
  float fm = -INFINITY; int fi = 0x7FFFFFFF;
  #pragma unroll
  for (int t = 0; t < 5; ++t) {
    int idx = lane + 32 * t;
    if (idx >= 1 && idx < CC && v[t] > fm) { fm = v[t]; fi = idx; }
  }
  waveArgMax(fm, fi);
  if (lane == 0) {
    outPred[row]  = (float)fi;            // already argmax(fg)+1
    outScore[row] = __expf(fm - m) / s;
  }
}

// ---------------------------------------------------------------------------
// Kernel B: stream rel_logits through LDS (async copy), build sort keys.
// key = (~sortable(triple) << 32) | r  -> ascending u64 == stable desc sort
// ---------------------------------------------------------------------------
__global__ void rel_key_kernel(const float* __restrict__ rel,
                               const long long* __restrict__ pairs,
                               const float* __restrict__ objScore,
                               u64* __restrict__ keys) {
  __shared__ __align__(16) float tile[ROWS_PER_BLK * PP];
  const int tid = threadIdx.x;
  const size_t firstRow = (size_t)blockIdx.x * ROWS_PER_BLK;   // global row index
  const char* gbase = (const char*)(rel + firstRow * PP);      // 16B aligned

#if HAVE_ASYNC
  for (int c = tid; c < CHUNKS; c += 256)
    async_cp16(gbase + (size_t)c * 16, (char*)tile + c * 16);
  wait_async0();
#else
  for (int c = tid; c < CHUNKS; c += 256)
    ((float4*)tile)[c] = ((const float4*)gbase)[c];
#endif
  __syncthreads();

  const int wave = tid >> 5, lane = tid & 31;
  #pragma unroll
  for (int it = 0; it < ROWS_PER_BLK / 8; ++it) {
    const int rl = wave + it * 8;
    const float* rowp = tile + rl * PP;
    float x0 = rowp[lane];
    float x1 = (lane + 32 < PP) ? rowp[lane + 32] : -INFINITY;
    float m = waveMax(fmaxf(x0, x1));
    float s = __expf(x0 - m) + ((lane + 32 < PP) ? __expf(x1 - m) : 0.f);
    s = waveSum(s);
    float fm = (lane >= 1) ? x0 : -INFINITY;          // exclude class 0
    if (lane + 32 < PP && x1 > fm) fm = x1;
    fm = waveMax(fm);
    if (lane == 0) {
      const size_t e = firstRow + rl;                 // = b*R + r
      const int b = (int)(e >> 16);
      const long long* pp2 = pairs + 2 * e;
      const int i0 = (int)pp2[0], i1 = (int)pp2[1];
      const float s0 = objScore[b * NN + i0];
      const float s1 = objScore[b * NN + i1];
      const float triple = (__expf(fm - m) / s) * s0 * s1;
      keys[e] = ((u64)desc_key(triple) << 32) | (u32)(e & (RR - 1));
    }
  }
}

// ---------------------------------------------------------------------------
// Hybrid bitonic sort of u64 keys, 16 independent 64K-element batches.
// Direction always computed from the element's index within its batch.
// ---------------------------------------------------------------------------
__global__ void bitonic_tile_sort(u64* __restrict__ keys) {    // k = 2..TILE in LDS
  __shared__ u64 sm[TILE];
  const u32 tid = threadIdx.x;                                  // 1024 threads
  const size_t base = (size_t)blockIdx.x * TILE;
  const u32 rbase = (u32)((blockIdx.x & (RR / TILE - 1)) * TILE);
  sm[tid] = keys[base + tid];
  sm[tid + 1024] = keys[base + tid + 1024];
  for (u32 k = 2; k <= TILE; k <<= 1) {
    for (u32 j = k >> 1; j > 0; j >>= 1) {
      __syncthreads();
      const u32 i = 2u * tid - (tid & (j - 1));
      const bool dir = (((rbase + i) & k) == 0);
      u64 a = sm[i], b = sm[i + j];
      if ((a > b) == dir) { sm[i] = b; sm[i + j] = a; }
    }
  }
  __syncthreads();
  keys[base + tid] = sm[tid];
  keys[base + tid + 1024] = sm[tid + 1024];
}

__global__ void bitonic_global_pass(u64* __restrict__ keys, u32 j, u32 k) {
  const u32 t  = blockIdx.x * blockDim.x + threadIdx.x;         // < B*R/2
  const u32 tr = t & (RR / 2 - 1);
  const size_t base = ((size_t)(t >> 15)) << 16;                // b * R
  const u32 i = 2u * tr - (tr & (j - 1));
  u64 a = keys[base + i], b = keys[base + i + j];
  const bool dir = ((i & k) == 0);
  if ((a > b) == dir) { keys[base + i] = b; keys[base + i + j] = a; }
}

__global__ void bitonic_tile_finish(u64* __restrict__ keys, u32 k) { // j<=TILE/2 in LDS
  __shared__ u64 sm[TILE];
  const u32 tid = threadIdx.x;
  const size_t base = (size_t)blockIdx.x * TILE;
  const u32 rbase = (u32)((blockIdx.x & (RR / TILE - 1)) * TILE);
  sm[tid] = keys[base + tid];
  sm[tid + 1024] = keys[base + tid + 1024];
  for (u32 j = TILE >> 1; j > 0; j >>= 1) {
    __syncthreads();
    const u32 i = 2u * tid - (tid & (j - 1));
    const bool dir = (((rbase + i) & k) == 0);
    u64 a = sm[i], b = sm[i + j];
    if ((a > b) == dir) { sm[i] = b; sm[i + j] = a; }
  }
  __syncthreads();
  keys[base + tid] = sm[tid];
  keys[base + tid + 1024] = sm[tid + 1024];
}

// ---------------------------------------------------------------------------
// Kernel D: gather. For sorted slot e = b*R + j: recompute full softmax of the
// winning row (per-batch 13.4MB slice is L2-resident) and emit all 4 sorted
// outputs. One wave per slot.
// ---------------------------------------------------------------------------
__global__ void gather_kernel(const float* __restrict__ rel,
                              const long long* __restrict__ pairs,
                              const float* __restrict__ objScore,
                              const u64* __restrict__ keys,
                              float* __restrict__ outPairs,
                              float* __restrict__ outProb,
                              float* __restrict__ outLabel,
                              float* __restrict__ outTriple) {
  const int wave = threadIdx.x >> 5, lane = threadIdx.x & 31;
  const size_t e = (size_t)blockIdx.x * 8 + wave;     // b*R + sorted position
  const int b = (int)(e >> 16);
  const u32 r = (u32)(keys[e] & (RR - 1));
  const long long* pp2 = pairs + 2 * ((size_t)b * RR + r);
  __builtin_prefetch(pp2, 0, 0);                      // global_prefetch_b8
  const float* rowp = rel + ((size_t)b * RR + r) * PP;
  float x0 = rowp[lane];
  float x1 = (lane + 32 < PP) ? rowp[lane + 32] : -INFINITY;
  float m  = waveMax(fmaxf(x0, x1));
  float e0 = __expf(x0 - m);
  float e1 = (lane + 32 < PP) ? __expf(x1 - m) : 0.f;
  float s  = waveSum(e0 + e1);
  float inv = 1.0f / s;
  float* pr = outProb + e * PP;
  pr[lane] = e0 * inv;
  if (lane + 32 < PP) pr[lane + 32] = e1 * inv;
  float fm = (lane >= 1) ? x0 : -INFINITY;
  int   fi = (lane >= 1) ? lane : 0x7FFFFFFF;
  if (lane + 32 < PP && x1 > fm) { fm = x1; fi = lane + 32; }
  waveArgMax(fm, fi);
  if (lane == 0) {
    const int i0 = (int)pp2[0], i1 = (int)pp2[1];
    outPairs[2 * e]     = (float)i0;
    outPairs[2 * e + 1] = (float)i1;
    outLabel[e]  = (float)fi;
    outTriple[e] = __expf(fm - m) * inv * objScore[b * NN + i0] * objScore[b * NN + i1];
  }
}

// ---------------------------------------------------------------------------
extern "C" void kernel_launch(void* const* d_in, const int* in_sizes, int n_in,
                              void* d_out, int out_size, void* d_ws, size_t ws_size,
                              hipStream_t stream) {
  const float* rel       = (const float*)d_in[0];        // (B,R,P) f32
  const float* obj       = (const float*)d_in[1];        // (B,N,C) f32
  const long long* pairs = (const long long*)d_in[2];    // (B,R,2) i64

  float* out       = (float*)d_out;
  float* outPred   = out;                                 // B*N
  float* outScore  = out + BB * NN;                       // B*N
  float* outPairs  = out + 2 * BB * NN;                   // B*R*2
  float* outProb   = outPairs + (size_t)BB * RR * 2;      // B*R*P
  float* outLabel  = outProb + (size_t)BB * RR * PP;      // B*R
  float* outTriple = outLabel + (size_t)BB * RR;          // B*R

  u64* keys = (u64*)d_ws;                                 // B*R*8 = 8 MB

  obj_softmax_kernel<<<BB * NN / 8, 256, 0, stream>>>(obj, outPred, outScore);
  rel_key_kernel<<<BB * RR / ROWS_PER_BLK, 256, 0, stream>>>(rel, pairs, outScore, keys);

  bitonic_tile_sort<<<(BB * RR) / TILE, 1024, 0, stream>>>(keys);
  for (u32 k = 2 * TILE; k <= RR; k <<= 1) {
    for (u32 j = k >> 1; j >= TILE; j >>= 1)
      bitonic_global_pass<<<(BB * RR / 2) / 256, 256, 0, stream>>>(keys, j, k);
    bitonic_tile_finish<<<(BB * RR) / TILE, 1024, 0, stream>>>(keys, k);
  }

  gather_kernel<<<BB * RR / 8, 256, 0, stream>>>(rel, pairs, outScore, keys,
                                                 outPairs, outProb, outLabel, outTriple);
}